// PointNetSetAbstraction_53412213293939
// MI455X (gfx1250) — compile-verified
//
#include <hip/hip_runtime.h>
#include <hip/hip_bf16.h>

// PointNet++ Set Abstraction for MI455X (gfx1250, wave32, WMMA).
// Pipeline: FPS -> ball query -> fused [gather + GEMM + bias + BN-stats] x3
// (BN-normalize+ReLU of layer k fused into LDS staging of layer k+1)
// -> BN-finalize -> max-pool over samples.

typedef float v2f __attribute__((ext_vector_type(2)));
typedef float v8f __attribute__((ext_vector_type(8)));

#define BATCH 8
#define NPTS  4096
#define NP    1024          // npoint
#define NS    32            // nsample
#define NCOLS (BATCH*NP*NS) // 262144 columns for the 1x1-conv GEMMs
#define CNTF  262144.0f     // BN reduction count = B*S*P
#define BN_EPS 1e-5f

// ---------------------------------------------------------------------------
// Farthest point sampling: one block per batch. dist kept in registers,
// block-wide argmax via LDS tree. Emits fps index + new_xyz (first output).
// Reference seeds the start index from the JAX PRNG; we use index 0
// (unreproducible PRNG; first centroid only).
// ---------------------------------------------------------------------------
__global__ __launch_bounds__(512) void fps_kernel(const float* __restrict__ xyz,
                                                  int* __restrict__ fpsIdx,
                                                  float* __restrict__ newXyz) {
  __shared__ float rv[512];
  __shared__ int   ri[512];
  __shared__ int   sFar;
  const int b   = blockIdx.x;
  const int tid = threadIdx.x;
  const float* xb = xyz + (size_t)b * NPTS * 3;

  float dloc[8];
#pragma unroll
  for (int j = 0; j < 8; ++j) dloc[j] = 1e10f;
  if (tid == 0) sFar = 0;
  __syncthreads();

  for (int it = 0; it < NP; ++it) {
    const int f = sFar;
    const float cx = xb[f*3+0], cy = xb[f*3+1], cz = xb[f*3+2];
    if (tid == 0) {
      fpsIdx[b*NP + it] = f;
      newXyz[(size_t)(b*NP + it)*3 + 0] = cx;
      newXyz[(size_t)(b*NP + it)*3 + 1] = cy;
      newXyz[(size_t)(b*NP + it)*3 + 2] = cz;
    }
    float bv = -1.0f; int bi = 0;
#pragma unroll
    for (int j = 0; j < 8; ++j) {
      const int p = j*512 + tid;
      const float dx = xb[p*3+0]-cx, dy = xb[p*3+1]-cy, dz = xb[p*3+2]-cz;
      const float d  = dx*dx + dy*dy + dz*dz;
      const float nd = fminf(dloc[j], d);
      dloc[j] = nd;
      if (nd > bv) { bv = nd; bi = p; }   // strict > keeps lowest index (argmax tie rule)
    }
    rv[tid] = bv; ri[tid] = bi;
    __syncthreads();
    for (int st = 256; st > 0; st >>= 1) {
      if (tid < st) {
        const float ov = rv[tid+st]; const int oi = ri[tid+st];
        if (ov > rv[tid] || (ov == rv[tid] && oi < ri[tid])) { rv[tid] = ov; ri[tid] = oi; }
      }
      __syncthreads();
    }
    if (tid == 0) sFar = ri[0];
    __syncthreads();
  }
}

// ---------------------------------------------------------------------------
// Ball query: one wave32 per centroid. Sequential scan gives ascending index
// order == reference's sort-then-truncate. Compaction via wave ballot.
// ---------------------------------------------------------------------------
__global__ __launch_bounds__(256) void ball_kernel(const float* __restrict__ xyz,
                                                   const float* __restrict__ newXyz,
                                                   int* __restrict__ ballIdx) {
  const int lane = threadIdx.x & 31;
  const int wv   = threadIdx.x >> 5;
  const int g    = blockIdx.x * 8 + wv;    // centroid 0..8191
  const int b    = g >> 10;
  const float* xb = xyz + (size_t)b * NPTS * 3;
  const float cx = newXyz[(size_t)g*3+0];
  const float cy = newXyz[(size_t)g*3+1];
  const float cz = newXyz[(size_t)g*3+2];
  int* out = ballIdx + (size_t)g * NS;

  int count = 0, first = 0; bool haveFirst = false;
  for (int base = 0; base < NPTS; base += 32) {
    const int i = base + lane;
    const float dx = xb[i*3+0]-cx, dy = xb[i*3+1]-cy, dz = xb[i*3+2]-cz;
    const bool in = (dx*dx + dy*dy + dz*dz) <= 1.0f;  // RADIUS^2
    const unsigned m = __builtin_amdgcn_ballot_w32(in);
    if (!haveFirst && m) { first = base + __builtin_ctz(m); haveFirst = true; }
    const int rank = __builtin_popcount(m & ((1u << lane) - 1u));
    const int pos  = count + rank;
    if (in && pos < NS) out[pos] = i;
    count += __builtin_popcount(m);
    if (count >= NS) break;               // uniform across wave
  }
  const int filled = count < NS ? count : NS;
  if (lane >= filled) out[lane] = first;  // pad with first neighbor
}

// ---------------------------------------------------------------------------
// Zero the BN accumulators (sum + sumsq, 3 layers x 128 channels).
// ---------------------------------------------------------------------------
__global__ void zero_stats_kernel(float* __restrict__ stats) {
  const int layer = blockIdx.x;           // 0..2
  stats[layer*512 + threadIdx.x] = 0.0f;  // 256 threads: sum[128] + sq[128]
}

// ---------------------------------------------------------------------------
// Fused GEMM: Y[oc, col] = W[oc,:] . X[:, col] + bias[oc]
//   MODE 0: X gathered on the fly (3 relative-xyz + 64 point features)
//   MODE 1: X = relu(Yprev*scaleP + shiftP)  (previous layer's BN+ReLU)
// Block: 128 threads (4 waves). Tile: 64 M x 64 N. Each wave: 16 M rows,
// four v_wmma_f32_16x16x4_f32 accumulators covering all 64 N columns.
// Epilogue accumulates per-channel BN sum/sumsq (LDS f32 atomics -> 1 global
// atomic per channel per block).
// ---------------------------------------------------------------------------
template <int IC, int KPAD, int MODE>
__global__ __launch_bounds__(128) void gemm_bn_kernel(
    const float* __restrict__ W, const float* __restrict__ bias,
    float* __restrict__ Y, float* __restrict__ gSum, float* __restrict__ gSq,
    const float* __restrict__ xyz, const float* __restrict__ points,
    const int* __restrict__ ballIdx, const float* __restrict__ newXyz,
    const float* __restrict__ Yprev, const float* __restrict__ scaleP,
    const float* __restrict__ shiftP) {
  __shared__ float Wl[64 * KPAD];   // weights tile  [64 oc][KPAD k]
  __shared__ float Xl[KPAD * 64];   // input tile    [KPAD k][64 col]
  __shared__ float sSum[64];
  __shared__ float sSq[64];

  const int tid   = threadIdx.x;
  const int n0    = blockIdx.x * 64;   // column base
  const int mBase = blockIdx.y * 64;   // output-channel base

  // --- stage weights (zero-pad K to KPAD) ---
  for (int L = tid; L < 64 * KPAD; L += 128) {
    const int oc = L / KPAD, k = L % KPAD;
    Wl[L] = (k < IC) ? W[(size_t)(mBase + oc) * IC + k] : 0.0f;
  }

  // --- stage input tile ---
  if (MODE == 0) {
    const int pbase = blockIdx.x * 2;               // 64 cols == 2 centroids
    for (int L = tid; L < KPAD * 64; L += 128) {
      const int c = L >> 6, j = L & 63;
      float v = 0.0f;
      if (c < IC) {
        const int pt = pbase + (j >> 5);
        const int s  = j & 31;
        const int bb = pt >> 10;
        const int id = ballIdx[(size_t)pt * NS + s];
        if (c < 3) v = xyz[((size_t)bb * NPTS + id) * 3 + c] - newXyz[(size_t)pt * 3 + c];
        else       v = points[((size_t)bb * NPTS + id) * 64 + (c - 3)];
      }
      Xl[c * 64 + j] = v;
    }
  } else {
    for (int L = tid; L < KPAD * 64; L += 128) {
      const int c = L >> 6, j = L & 63;
      float v = 0.0f;
      if (c < IC) {
        const float y = Yprev[(size_t)c * NCOLS + n0 + j];
        v = fmaxf(y * scaleP[c] + shiftP[c], 0.0f);   // fused BN + ReLU
      }
      Xl[c * 64 + j] = v;
    }
  }
  if (tid < 64) { sSum[tid] = 0.0f; sSq[tid] = 0.0f; }
  __syncthreads();

  const int lane = tid & 31;
  const int wv   = tid >> 5;
  const int mW   = wv * 16;       // wave's M offset within the 64-row tile
  const int half = lane >> 4;     // 0: lanes 0-15, 1: lanes 16-31
  const int l15  = lane & 15;

  v8f acc[4];
  const v8f vzero = {0.f, 0.f, 0.f, 0.f, 0.f, 0.f, 0.f, 0.f};
#pragma unroll
  for (int nt = 0; nt < 4; ++nt) acc[nt] = vzero;

  // ISA 7.12.2 fp32 16x4 A layout: lanes 0-15 hold K={k0,k0+1}, 16-31 hold {k0+2,k0+3}
#pragma unroll 4
  for (int k0 = 0; k0 < KPAD; k0 += 4) {
    v2f a;
    a.x = Wl[(mW + l15) * KPAD + k0 + 2 * half];
    a.y = Wl[(mW + l15) * KPAD + k0 + 2 * half + 1];
#pragma unroll
    for (int nt = 0; nt < 4; ++nt) {
      v2f bf;
      bf.x = Xl[(k0 + 2 * half)     * 64 + nt * 16 + l15];
      bf.y = Xl[(k0 + 2 * half + 1) * 64 + nt * 16 + l15];
      acc[nt] = __builtin_amdgcn_wmma_f32_16x16x4_f32(
          false, a, false, bf, (short)0, acc[nt], false, false);
    }
  }

  // --- epilogue: bias, store, BN statistics ---
  float ls[8], lq[8];
#pragma unroll
  for (int i = 0; i < 8; ++i) { ls[i] = 0.0f; lq[i] = 0.0f; }
#pragma unroll
  for (int nt = 0; nt < 4; ++nt) {
#pragma unroll
    for (int i = 0; i < 8; ++i) {
      const int ocl = mW + i + 8 * half;      // C/D layout: vgpr i -> M = i (+8 hi-half)
      const int oc  = mBase + ocl;
      const int col = n0 + nt * 16 + l15;
      const float v = acc[nt][i] + bias[oc];
      Y[(size_t)oc * NCOLS + col] = v;
      ls[i] += v; lq[i] += v * v;
    }
  }
#pragma unroll
  for (int i = 0; i < 8; ++i) {
    const int ocl = mW + i + 8 * half;
    atomicAdd(&sSum[ocl], ls[i]);
    atomicAdd(&sSq[ocl],  lq[i]);
  }
  __syncthreads();
  if (tid < 64) {
    atomicAdd(&gSum[mBase + tid], sSum[tid]);
    atomicAdd(&gSq[mBase + tid],  sSq[tid]);
  }
}

// ---------------------------------------------------------------------------
// BN finalize: scale = gamma*rsqrt(var+eps), shift = beta - mean*scale
// ---------------------------------------------------------------------------
__global__ void finalize_kernel(const float* __restrict__ sum,
                                const float* __restrict__ sq,
                                const float* __restrict__ gamma,
                                const float* __restrict__ beta,
                                float* __restrict__ scale,
                                float* __restrict__ shift, int OC) {
  const int i = threadIdx.x;
  if (i < OC) {
    const float mean = sum[i] / CNTF;
    const float var  = sq[i] / CNTF - mean * mean;
    const float sc   = gamma[i] * rsqrtf(var + BN_EPS);
    scale[i] = sc;
    shift[i] = beta[i] - mean * sc;
  }
}

// ---------------------------------------------------------------------------
// Max-pool over the 32 samples of each centroid. One block per centroid,
// one thread per output channel: each thread streams one 128-byte line.
// ---------------------------------------------------------------------------
__global__ __launch_bounds__(128) void maxpool_kernel(const float* __restrict__ Y3,
                                                      const float* __restrict__ scale,
                                                      const float* __restrict__ shift,
                                                      float* __restrict__ outPts) {
  const int pt = blockIdx.x;     // 0..8191
  const int oc = threadIdx.x;    // 0..127
  const float* row = Y3 + (size_t)oc * NCOLS + (size_t)pt * NS;
  const float sc = scale[oc], sh = shift[oc];
  float m = -1e30f;
#pragma unroll
  for (int s = 0; s < NS; ++s) m = fmaxf(m, row[s] * sc + sh);
  m = fmaxf(m, 0.0f);            // ReLU commutes with max
  outPts[(size_t)pt * 128 + oc] = m;
}

// ---------------------------------------------------------------------------
extern "C" void kernel_launch(void* const* d_in, const int* in_sizes, int n_in,
                              void* d_out, int out_size, void* d_ws, size_t ws_size,
                              hipStream_t stream) {
  (void)in_sizes; (void)n_in; (void)out_size; (void)ws_size;
  const float* xyz    = (const float*)d_in[0];
  const float* points = (const float*)d_in[1];
  const float* W0 = (const float*)d_in[2];  const float* b0  = (const float*)d_in[3];
  const float* g0 = (const float*)d_in[4];  const float* be0 = (const float*)d_in[5];
  const float* W1 = (const float*)d_in[6];  const float* b1  = (const float*)d_in[7];
  const float* g1 = (const float*)d_in[8];  const float* be1 = (const float*)d_in[9];
  const float* W2 = (const float*)d_in[10]; const float* b2  = (const float*)d_in[11];
  const float* g2 = (const float*)d_in[12]; const float* be2 = (const float*)d_in[13];

  float* out    = (float*)d_out;
  float* newXyz = out;             // (8,1024,3)  = 24576 floats
  float* newPts = out + 24576;     // (8,1024,128)

  char* ws = (char*)d_ws;
  float* Y1 = (float*)(ws);                                   //  64 x NCOLS
  float* Y2 = (float*)(ws + (size_t)64  * NCOLS * 4);         //  64 x NCOLS
  float* Y3 = (float*)(ws + (size_t)128 * NCOLS * 4);         // 128 x NCOLS
  int*   ballIdx = (int*)(ws + (size_t)256 * NCOLS * 4);      // 8*1024*32 ints
  int*   fpsIdx  = (int*)(ws + (size_t)256 * NCOLS * 4 + (size_t)NCOLS * 4);
  float* stats   = (float*)(ws + (size_t)256 * NCOLS * 4 + (size_t)NCOLS * 4 + 8192 * 4);
  // per layer: sum[128], sq[128], scale[128], shift[128]
  float* sum0 = stats +    0; float* sq0 = stats +  128;
  float* sc0  = stats +  256; float* sh0 = stats +  384;
  float* sum1 = stats +  512; float* sq1 = stats +  640;
  float* sc1  = stats +  768; float* sh1 = stats +  896;
  float* sum2 = stats + 1024; float* sq2 = stats + 1152;
  float* sc2  = stats + 1280; float* sh2 = stats + 1408;

  fps_kernel<<<BATCH, 512, 0, stream>>>(xyz, fpsIdx, newXyz);
  ball_kernel<<<(BATCH * NP) / 8, 256, 0, stream>>>(xyz, newXyz, ballIdx);
  zero_stats_kernel<<<3, 256, 0, stream>>>(stats);

  // Layer 0: 67 -> 64 (gather fused), K padded 67->68
  gemm_bn_kernel<67, 68, 0><<<dim3(NCOLS / 64, 1), 128, 0, stream>>>(
      W0, b0, Y1, sum0, sq0, xyz, points, ballIdx, newXyz, nullptr, nullptr, nullptr);
  finalize_kernel<<<1, 128, 0, stream>>>(sum0, sq0, g0, be0, sc0, sh0, 64);

  // Layer 1: 64 -> 64 (BN+ReLU of layer 0 fused into load)
  gemm_bn_kernel<64, 64, 1><<<dim3(NCOLS / 64, 1), 128, 0, stream>>>(
      W1, b1, Y2, sum1, sq1, nullptr, nullptr, nullptr, nullptr, Y1, sc0, sh0);
  finalize_kernel<<<1, 128, 0, stream>>>(sum1, sq1, g1, be1, sc1, sh1, 64);

  // Layer 2: 64 -> 128
  gemm_bn_kernel<64, 64, 1><<<dim3(NCOLS / 64, 2), 128, 0, stream>>>(
      W2, b2, Y3, sum2, sq2, nullptr, nullptr, nullptr, nullptr, Y2, sc1, sh1);
  finalize_kernel<<<1, 128, 0, stream>>>(sum2, sq2, g2, be2, sc2, sh2, 128);

  // BN+ReLU of layer 2 fused into max-pool
  maxpool_kernel<<<BATCH * NP, 128, 0, stream>>>(Y3, sc2, sh2, newPts);
}